// ManyToManyMultiheadMonotonicAttention_32590211842719
// MI455X (gfx1250) — compile-verified
//
#include <hip/hip_runtime.h>
#include <hip/hip_bf16.h>

typedef __attribute__((ext_vector_type(2))) float v2f;
typedef __attribute__((ext_vector_type(8))) float v8f;

#define EMBED 1024
#define TQ 1024
#define HD 64
#define EPS_F 0.001f

// D = A(16x4) x B(4x16) + C, all f32, wave32.
__device__ __forceinline__ v8f wmma4(v2f a, v2f b, v8f c) {
  return __builtin_amdgcn_wmma_f32_16x16x4_f32(
      /*neg_a=*/false, a, /*neg_b=*/false, b,
      /*c_mod=*/(short)0, c, /*reuse_a=*/false, /*reuse_b=*/false);
}

// ---------------- GEMM: C(M,N) = X(M,K) @ W(N,K)^T + bias(N) ----------------
// Block = 256 threads = 8 waves arranged 4(M) x 2(N); wave strip = 16 x 64.
// K staged in 16-wide chunks, double-buffered in registers so loads for chunk
// i+1 are in flight while chunk i's WMMAs run (partial s_wait_loadcnt).

struct GemmChunk {
  v2f a[4];
  v2f b[4][4];
};

__device__ __forceinline__ void gemm_load_chunk(
    GemmChunk& c, const float* __restrict__ xrow,
    const float* __restrict__ w0, const float* __restrict__ w1,
    const float* __restrict__ w2, const float* __restrict__ w3, int k) {
#pragma unroll
  for (int s = 0; s < 4; ++s) {
    const int ks = k + 4 * s;
    c.a[s]    = *(const v2f*)(xrow + ks);
    c.b[s][0] = *(const v2f*)(w0 + ks);
    c.b[s][1] = *(const v2f*)(w1 + ks);
    c.b[s][2] = *(const v2f*)(w2 + ks);
    c.b[s][3] = *(const v2f*)(w3 + ks);
  }
}

__device__ __forceinline__ void gemm_mma_chunk(v8f acc[4], const GemmChunk& c) {
#pragma unroll
  for (int s = 0; s < 4; ++s)
#pragma unroll
    for (int t = 0; t < 4; ++t)
      acc[t] = wmma4(c.a[s], c.b[s][t], acc[t]);
}

__global__ __launch_bounds__(256) void gemm_xwt_bias(
    const float* __restrict__ X, const float* __restrict__ W,
    const float* __restrict__ bias, float* __restrict__ C,
    int M, int N, int K) {
  const int lane = threadIdx.x & 31;
  const int wv = threadIdx.x >> 5;
  const int wm = wv & 3, wn = wv >> 2;
  const int m0 = blockIdx.y * 64 + wm * 16;
  const int n0 = blockIdx.x * 128 + wn * 64;
  const int mr = lane & 15;        // A row / B col within tile
  const int kh = (lane >> 4) * 2;  // k sub-offset per half-wave

  const float* xrow = X + (size_t)(m0 + mr) * K + kh;
  const float* w0 = W + (size_t)(n0 + 0 + mr) * K + kh;
  const float* w1 = W + (size_t)(n0 + 16 + mr) * K + kh;
  const float* w2 = W + (size_t)(n0 + 32 + mr) * K + kh;
  const float* w3 = W + (size_t)(n0 + 48 + mr) * K + kh;

  v8f acc[4];
#pragma unroll
  for (int t = 0; t < 4; ++t) acc[t] = (v8f){0, 0, 0, 0, 0, 0, 0, 0};

  GemmChunk c0, c1;
  gemm_load_chunk(c0, xrow, w0, w1, w2, w3, 0);
  for (int kc = 0; kc < K; kc += 32) {             // K % 32 == 0
    gemm_load_chunk(c1, xrow, w0, w1, w2, w3, kc + 16);
    gemm_mma_chunk(acc, c0);
    if (kc + 32 < K) gemm_load_chunk(c0, xrow, w0, w1, w2, w3, kc + 32);
    gemm_mma_chunk(acc, c1);
  }

  const int mrow = m0 + (lane >> 4) * 8;
#pragma unroll
  for (int t = 0; t < 4; ++t) {
    const int col = n0 + t * 16 + mr;
    const float bv = bias[col];
#pragma unroll
    for (int v = 0; v < 8; ++v)
      C[(size_t)(mrow + v) * N + col] = acc[t][v] + bv;
  }
}

// ---------------- Fused monotonic attention ----------------
// One workgroup (1024 threads = 32 wave32) per (b,g,h) head instance.

struct QKChunk {
  v2f a[4];
  v2f b[4];
};

__device__ __forceinline__ void qk_load_chunk(
    QKChunk& c, const float* __restrict__ arow, const float* __restrict__ brow,
    int k) {
#pragma unroll
  for (int s = 0; s < 4; ++s) {
    c.a[s] = *(const v2f*)(arow + k + 4 * s);
    c.b[s] = *(const v2f*)(brow + k + 4 * s);
  }
}

__device__ __forceinline__ void qk_mma_chunk(v8f& acc, const QKChunk& c) {
#pragma unroll
  for (int s = 0; s < 4; ++s) acc = wmma4(c.a[s], c.b[s], acc);
}

__global__ __launch_bounds__(1024) void mono_attn(
    const float* __restrict__ xq, const float* __restrict__ xk,
    const float* __restrict__ xv, float* __restrict__ xo) {
  extern __shared__ float smem[];
  float* p_blk   = smem;                  // 16*1024 : clipped sigmoid probs
  float* phi_blk = p_blk + 16 * TQ;       // 16*1024 : phi rows of this block
  float* p_last  = phi_blk + 16 * TQ;     // 1024    : p row i0-1 carried over
  float* scanA   = p_last + TQ;           // 2*1024  : scan ping-pong (A)
  float* scanB   = scanA + 2 * TQ;        // 2*1024  : scan ping-pong (b)
  float* parts   = scanB + 2 * TQ;        // 32*32*8 : split-K partial D tiles

  const int tid = threadIdx.x;
  const int lane = tid & 31;
  const int wv = tid >> 5;            // 0..31
  const int mr = lane & 15;
  const int kh = (lane >> 4) * 2;

  const int bid = blockIdx.x;         // 0..31
  const int b = bid >> 4;
  const int g = (bid >> 2) & 3;
  const int h = bid & 3;

  const float* Qh = xq + (size_t)b * TQ * EMBED + g * 256 + h * 64; // ld 1024
  const float* Kh = xk + (size_t)b * TQ * 256 + h * 64;             // ld 256
  const float* Vh = xv + (size_t)b * TQ * 256 + h * 64;             // ld 256
  float*       Oh = xo + (size_t)b * TQ * EMBED + g * 256 + h * 64; // ld 1024

  float phiPrev = 0.0f;  // thread j's phi[i-1, j]; set at first row

  for (int kb = 0; kb < TQ / 16; ++kb) {
    const int i0 = kb * 16;

    // ---- Phase 1: p_blk[16][1024] = clip(sigmoid(Q_blk @ K^T)) via WMMA ----
#pragma unroll
    for (int s = 0; s < 2; ++s) {
      const int j0 = (wv * 2 + s) * 16;
      const float* arow = Qh + (size_t)(i0 + mr) * EMBED + kh;
      const float* brow = Kh + (size_t)(j0 + mr) * 256 + kh;
      v8f acc = {0, 0, 0, 0, 0, 0, 0, 0};
      QKChunk q0, q1;                      // 4-chunk software pipeline (HD=64)
      qk_load_chunk(q0, arow, brow, 0);
      qk_load_chunk(q1, arow, brow, 16);
      qk_mma_chunk(acc, q0);
      qk_load_chunk(q0, arow, brow, 32);
      qk_mma_chunk(acc, q1);
      qk_load_chunk(q1, arow, brow, 48);
      qk_mma_chunk(acc, q0);
      qk_mma_chunk(acc, q1);

      const int mbase = (lane >> 4) * 8;
#pragma unroll
      for (int v = 0; v < 8; ++v) {
        float sg = 1.0f / (1.0f + __expf(-acc[v]));
        sg = fminf(fmaxf(sg, EPS_F), 1.0f - EPS_F);
        p_blk[(mbase + v) * TQ + j0 + mr] = sg;
      }
    }
    __syncthreads();

    // ---- Phase 2: serial rows, parallel affine scan over keys ----
    const int j = tid;
    for (int r = 0; r < 16; ++r) {
      float phi;
      if (kb == 0 && r == 0) {
        phi = (j == 0) ? 1.0f : 0.0f;     // phi[0] = onehot(key 0)
      } else {
        const float* pprev = (r == 0) ? p_last : (p_blk + (r - 1) * TQ);
        const float* pcur  = p_blk + r * TQ;
        const float aj = phiPrev * pprev[j];
        const float Aj = (j == 0) ? 0.0f : (1.0f - pcur[j - 1]);
        scanA[j] = Aj;
        scanB[j] = aj;
        __syncthreads();
        int buf = 0;
#pragma unroll
        for (int d = 1; d < TQ; d <<= 1) {
          const float A2 = scanA[buf * TQ + j];
          const float b2 = scanB[buf * TQ + j];
          float A1 = 1.0f, b1 = 0.0f;     // identity for j < d
          if (j >= d) {
            A1 = scanA[buf * TQ + j - d];
            b1 = scanB[buf * TQ + j - d];
          }
          buf ^= 1;
          scanA[buf * TQ + j] = A1 * A2;
          scanB[buf * TQ + j] = fmaf(b1, A2, b2);
          __syncthreads();
        }
        phi = scanB[buf * TQ + j];        // buf back to 0 after 10 steps
      }
      phi_blk[r * TQ + j] = phi;
      phiPrev = phi;
    }
    __syncthreads();

    // ---- Phase 3: O_blk = phi_blk(16x1024) @ V(1024x64), split-K WMMA ----
    {
      const int tile = wv & 3;            // output column tile (c0 = 16*tile)
      const int kseg = wv >> 2;           // 8 K-segments of 128
      const int c0 = tile * 16;
      const int kbeg = kseg * 128;
      const float* phiRow = phi_blk + mr * TQ + kh;
      const float* vcol = Vh + c0 + mr;

      QKChunk p0, p1;                      // a from LDS, b = strided V columns
      auto pv_load = [&](QKChunk& c, int k) {
#pragma unroll
        for (int s = 0; s < 4; ++s) {
          const int kr = k + 4 * s;
          c.a[s] = *(const v2f*)(phiRow + kr);
          v2f bb;
          bb.x = vcol[(size_t)(kr + kh) * 256];
          bb.y = vcol[(size_t)(kr + kh + 1) * 256];
          c.b[s] = bb;
        }
      };
      v8f acc = {0, 0, 0, 0, 0, 0, 0, 0};
      pv_load(p0, kbeg);
      for (int k0 = kbeg; k0 < kbeg + 128; k0 += 32) {
        pv_load(p1, k0 + 16);
        qk_mma_chunk(acc, p0);
        if (k0 + 32 < kbeg + 128) pv_load(p0, k0 + 32);
        qk_mma_chunk(acc, p1);
      }
#pragma unroll
      for (int v = 0; v < 8; ++v)
        parts[(wv * 32 + lane) * 8 + v] = acc[v];
    }
    __syncthreads();
    if (wv < 4) {                          // reduce 8 K-segment partials
      float sum[8] = {0, 0, 0, 0, 0, 0, 0, 0};
#pragma unroll
      for (int seg = 0; seg < 8; ++seg)
#pragma unroll
        for (int v = 0; v < 8; ++v)
          sum[v] += parts[((seg * 4 + wv) * 32 + lane) * 8 + v];
      const int mbase = (lane >> 4) * 8;
#pragma unroll
      for (int v = 0; v < 8; ++v)
        Oh[(size_t)(i0 + mbase + v) * EMBED + wv * 16 + mr] = sum[v];
    }
    __syncthreads();
    p_last[tid] = p_blk[15 * TQ + tid];   // carry p[i0+15] to next block
    __syncthreads();
  }
}

extern "C" void kernel_launch(void* const* d_in, const int* in_sizes, int n_in,
                              void* d_out, int out_size, void* d_ws, size_t ws_size,
                              hipStream_t stream) {
  const float* query = (const float*)d_in[0];
  const float* key   = (const float*)d_in[1];
  const float* value = (const float*)d_in[2];
  const float* w_in  = (const float*)d_in[3];
  const float* b_in  = (const float*)d_in[4];
  const float* w_out = (const float*)d_in[5];
  const float* b_out = (const float*)d_in[6];
  float* out = (float*)d_out;

  const int M = 2 * TQ;  // B*T = 2048 token rows
  float* xq = (float*)d_ws;                 // 2048 x 1024
  float* xk = xq + (size_t)M * EMBED;       // 2048 x 256
  float* xv = xk + (size_t)M * 256;         // 2048 x 256
  float* xo = xv + (size_t)M * 256;         // 2048 x 1024  (total ~20 MB)

  dim3 blk(256);
  // Input projections (W is (N,K) row-major; rows: Q[0:1024), K[1024:1280), V[1280:1536))
  gemm_xwt_bias<<<dim3(EMBED / 128, M / 64), blk, 0, stream>>>(
      query, w_in, b_in, xq, M, EMBED, EMBED);
  gemm_xwt_bias<<<dim3(256 / 128, M / 64), blk, 0, stream>>>(
      key, w_in + (size_t)EMBED * EMBED, b_in + EMBED, xk, M, 256, EMBED);
  gemm_xwt_bias<<<dim3(256 / 128, M / 64), blk, 0, stream>>>(
      value, w_in + (size_t)(EMBED + 256) * EMBED, b_in + EMBED + 256, xv, M, 256, EMBED);

  // Fused monotonic attention: 32 head instances, one 32-wave WG each.
  const size_t smem =
      (size_t)(16 * TQ + 16 * TQ + TQ + 2 * TQ + 2 * TQ + 32 * 32 * 8) * sizeof(float);
  mono_attn<<<32, 1024, smem, stream>>>(xq, xk, xv, xo);

  // Output projection
  gemm_xwt_bias<<<dim3(EMBED / 128, M / 64), blk, 0, stream>>>(
      xo, w_out, b_out, out, M, EMBED, EMBED);
}